// InsectAwareProtoPool_1700807049524
// MI455X (gfx1250) — compile-verified
//
#include <hip/hip_runtime.h>
#include <stdint.h>

// Problem constants (from reference).
#define DIM        768
#define N_STAGES   4
#define MAXP       8
#define SHP        16
#define NCHUNK     (DIM / 4)   // 192 float4 chunks per row -> 192 threads / block

// ---------------------------------------------------------------------------
// gfx1250 async global->LDS copy (ASYNCcnt-tracked). Inline asm so it works on
// both ROCm 7.2 and the amdgpu-toolchain lane (bypasses builtin arity skew).
// GV addressing mode: 64-bit vaddr, saddr = off. First operand is the LDS
// byte-address VGPR (VDST), second is the global address VGPR pair.
// ---------------------------------------------------------------------------
__device__ __forceinline__ void async_copy_b128_to_lds(uint32_t lds_byte_addr,
                                                       const float* gptr) {
    asm volatile("global_load_async_to_lds_b128 %0, %1, off"
                 :
                 : "v"(lds_byte_addr), "v"((uint64_t)(uintptr_t)gptr)
                 : "memory");
}

__device__ __forceinline__ void wait_async_zero() {
#if __has_builtin(__builtin_amdgcn_s_wait_asynccnt)
    __builtin_amdgcn_s_wait_asynccnt(0);
#else
    asm volatile("s_wait_asynccnt 0x0" ::: "memory");
#endif
}

// ---------------------------------------------------------------------------
// Kernel 1: shared_mean[s][d] = mean_p shared_protos[s][p][d]   (tiny)
// ---------------------------------------------------------------------------
__global__ void proto_shared_mean_kernel(const float* __restrict__ shared_protos,
                                         float* __restrict__ shared_mean) {
    const int idx = blockIdx.x * blockDim.x + threadIdx.x;
    if (idx >= N_STAGES * DIM) return;
    const int s = idx / DIM;
    const int d = idx - s * DIM;
    const float* p = shared_protos + (size_t)s * SHP * DIM + d;
    float acc = 0.f;
#pragma unroll
    for (int i = 0; i < SHP; ++i) acc += p[i * DIM];
    shared_mean[idx] = acc * (1.0f / (float)SHP);
}

// ---------------------------------------------------------------------------
// Kernel 2: one block per sample. Async-stage the valid proto rows into LDS
// (skipping masked rows halves the gather traffic vs the reference), overlap
// with the features/shared loads, then reduce and store.
// ---------------------------------------------------------------------------
__global__ void __launch_bounds__(NCHUNK)
proto_pool_main_kernel(const float* __restrict__ features,
                       const float* __restrict__ class_protos,
                       const float* __restrict__ shared_mean,
                       const int* __restrict__ class_ids,
                       const int* __restrict__ stages,
                       const int* __restrict__ proto_counts,
                       float* __restrict__ out) {
    __shared__ __align__(16) float smem[MAXP * DIM];   // 24 KB tile

    const int b   = blockIdx.x;       // sample id (block-uniform)
    const int tid = threadIdx.x;      // 0..191
    const int c   = tid * 4;          // float index of this thread's chunk

    // Block-uniform metadata -> scalarized by the compiler.
    const int cid = class_ids[b];
    const int st  = stages[b];
    int cnt = proto_counts[cid * N_STAGES + st];
    cnt = min(max(cnt, 0), MAXP);

    const float* base =
        class_protos + (size_t)(cid * N_STAGES + st) * (MAXP * DIM) + c;

    // Launch the gather: cnt async B128 copies per thread, zero VGPR landing
    // slots, tracked on ASYNCcnt.
    const uint32_t lds0 =
        (uint32_t)(uintptr_t)(&smem[0]) + (uint32_t)(c * (int)sizeof(float));
    for (int p = 0; p < cnt; ++p) {
        async_copy_b128_to_lds(lds0 + (uint32_t)(p * DIM * (int)sizeof(float)),
                               base + p * DIM);
    }

    // Overlap: pull features + shared chunk while the async DMA is in flight.
    const float4 f  = *(const float4*)(features    + (size_t)b * DIM + c);
    const float4 sh = *(const float4*)(shared_mean + st * DIM + c);

    wait_async_zero();   // each thread consumes exactly the bytes it staged

    float ax = 0.f, ay = 0.f, az = 0.f, aw = 0.f;
    const float* s0 = &smem[c];
    for (int p = 0; p < cnt; ++p) {
        const float4 v = *(const float4*)(s0 + p * DIM);
        ax += v.x; ay += v.y; az += v.z; aw += v.w;
    }

    const float w = 0.5f / (float)(cnt > 0 ? cnt : 1);   // 0.5 * 1/max(cnt,1)
    float4 o;
    o.x = fmaf(w, ax, fmaf(0.5f, sh.x, f.x));
    o.y = fmaf(w, ay, fmaf(0.5f, sh.y, f.y));
    o.z = fmaf(w, az, fmaf(0.5f, sh.z, f.z));
    o.w = fmaf(w, aw, fmaf(0.5f, sh.w, f.w));
    *(float4*)(out + (size_t)b * DIM + c) = o;
}

// ---------------------------------------------------------------------------
extern "C" void kernel_launch(void* const* d_in, const int* in_sizes, int n_in,
                              void* d_out, int out_size, void* d_ws, size_t ws_size,
                              hipStream_t stream) {
    const float* features      = (const float*)d_in[0];
    const float* class_protos  = (const float*)d_in[1];
    const float* shared_protos = (const float*)d_in[2];
    const int*   class_ids     = (const int*)d_in[3];
    const int*   stages        = (const int*)d_in[4];
    const int*   proto_counts  = (const int*)d_in[5];
    float*       out           = (float*)d_out;
    float*       shared_mean   = (float*)d_ws;   // 4*768 f32 = 12 KB scratch

    const int B = in_sizes[0] / DIM;   // 4096

    proto_shared_mean_kernel<<<(N_STAGES * DIM + 255) / 256, 256, 0, stream>>>(
        shared_protos, shared_mean);

    proto_pool_main_kernel<<<B, NCHUNK, 0, stream>>>(
        features, class_protos, shared_mean, class_ids, stages, proto_counts, out);
}